// Regressor_5007931867571
// MI455X (gfx1250) — compile-verified
//
#include <hip/hip_runtime.h>
#include <hip/hip_bf16.h>

typedef float v2f __attribute__((ext_vector_type(2)));
typedef float v8f __attribute__((ext_vector_type(8)));

#define NUM_GRAPHS 256

// ---------------------------------------------------------------------------
// Kernel 1: in-degree count (shared by both aggregation layers; same edges)
// ---------------------------------------------------------------------------
__global__ __launch_bounds__(256) void degree_count(const int* __restrict__ dst,
                                                    float* __restrict__ deg, int E) {
    int t = blockIdx.x * blockDim.x + threadIdx.x;
    if (t < E) atomicAdd(&deg[dst[t]], 1.0f);
}

// ---------------------------------------------------------------------------
// Kernel 2: edge scatter-add (copy_src + sum). Chunk-parallel: consecutive
// threads cover consecutive float4 chunks of one edge's feature row, so both
// the gather and the atomic destinations are contiguous within a wave.
// Feature tables fit in the 192MB L2, so gathers are L2 hits.
// ---------------------------------------------------------------------------
template <int F>
__global__ __launch_bounds__(256) void edge_scatter_add(const float* __restrict__ h,
                                                        const int* __restrict__ src,
                                                        const int* __restrict__ dst,
                                                        float* __restrict__ agg,
                                                        long long total) {
    constexpr int CH = F / 4;                 // float4 chunks per row
    constexpr int SH = (F == 128) ? 5 : 6;    // log2(CH)
    static_assert((1 << SH) == CH, "F must be 128 or 256");
    long long t = (long long)blockIdx.x * blockDim.x + threadIdx.x;
    if (t >= total) return;
    const long long e = t >> SH;
    const int c = (int)(t & (CH - 1));
    const int sN = src[e];
    const int dN = dst[e];
    const float4 v = *reinterpret_cast<const float4*>(h + (size_t)sN * F + c * 4);
    float* o = agg + (size_t)dN * F + c * 4;
    atomicAdd(o + 0, v.x);
    atomicAdd(o + 1, v.y);
    atomicAdd(o + 2, v.z);
    atomicAdd(o + 3, v.w);
}

// ---------------------------------------------------------------------------
// Kernel 3: fused (agg/deg) @ W + b, ReLU  -- exact-f32 WMMA 16x16x4.
// One wave per 16x16 output tile; 8 waves (256 threads) per block.
// nrows (=50000) is a multiple of 16, ncols fixed at 256 -> EXEC all-ones.
//
// A-operand layout (32-bit 16x4): lane<16 holds A[m=lane][k..k+1],
// lane>=16 holds A[m=lane-16][k+2..k+3]. B mirrors with columns.
// C/D: VGPR i -> row (lane<16 ? i : i+8), col = lane&15.
// ---------------------------------------------------------------------------
template <int K>
__global__ __launch_bounds__(256) void gemm_mean_bias_relu(const float* __restrict__ in,
                                                           const float* __restrict__ deg,
                                                           const float* __restrict__ W,
                                                           const float* __restrict__ bias,
                                                           float* __restrict__ out,
                                                           int nrows) {
    const int lane = threadIdx.x & 31;
    const int wave = threadIdx.x >> 5;
    const int tile = blockIdx.x * 8 + wave;
    const int tm = tile >> 4;            // 16 tiles along N-dim (256/16)
    const int tn = tile & 15;
    if (tm * 16 >= nrows) return;        // wave-uniform; EXEC stays full

    const int r  = lane & 15;            // row-in-tile for A, col-in-tile for B
    const int kh = (lane >> 4) << 1;     // 0 or 2: lane-half K offset

    const int m = tm * 16 + r;
    const int n = tn * 16 + r;
    const float s = 1.0f / fmaxf(deg[m], 1.0f);   // fused mean division

    const float* arow = in + (size_t)m * K;
    const float* bcol = W + n;

    v8f acc = {0.f, 0.f, 0.f, 0.f, 0.f, 0.f, 0.f, 0.f};
#pragma unroll
    for (int k = 0; k < K; k += 4) {
        v2f a, b;
        a.x = arow[k + kh] * s;
        a.y = arow[k + kh + 1] * s;
        b.x = bcol[(size_t)(k + kh) * 256];
        b.y = bcol[(size_t)(k + kh + 1) * 256];
        acc = __builtin_amdgcn_wmma_f32_16x16x4_f32(
            /*neg_a=*/false, a, /*neg_b=*/false, b,
            /*c_mod=*/(short)0, acc, /*reuse_a=*/false, /*reuse_b=*/false);
    }

    const float bn = bias[n];
    const int row0 = tm * 16 + ((lane >> 4) << 3);   // +0 or +8
#pragma unroll
    for (int i = 0; i < 8; ++i) {
        float v = fmaxf(acc[i] + bn, 0.0f);
        out[(size_t)(row0 + i) * 256 + n] = v;       // contiguous 64B per half-wave
    }
}

// ---------------------------------------------------------------------------
// Kernel 4: per-graph sum + count (mean_nodes numerator/denominator)
// ---------------------------------------------------------------------------
__global__ __launch_bounds__(256) void graph_accumulate(const float* __restrict__ h2,
                                                        const int* __restrict__ gid,
                                                        float* __restrict__ gsum,
                                                        float* __restrict__ gcnt,
                                                        long long total) {
    long long t = (long long)blockIdx.x * blockDim.x + threadIdx.x;
    if (t >= total) return;
    const int nidx = (int)(t >> 6);      // 256 feats / 4 = 64 chunks per node
    const int c = (int)(t & 63);
    const int g = gid[nidx];
    const float4 v = *reinterpret_cast<const float4*>(h2 + (size_t)nidx * 256 + c * 4);
    float* o = gsum + (size_t)g * 256 + c * 4;
    atomicAdd(o + 0, v.x);
    atomicAdd(o + 1, v.y);
    atomicAdd(o + 2, v.z);
    atomicAdd(o + 3, v.w);
    if (c == 0) atomicAdd(&gcnt[g], 1.0f);
}

// ---------------------------------------------------------------------------
// Kernel 5: readout MLP head. One block per graph, 128 threads (=H2).
// hidden[j] = br1[j] + sum_k (gsum[g][k]/cnt) * Wr1[k][j]; out = hidden.Wr2+br2
// ---------------------------------------------------------------------------
__global__ __launch_bounds__(128) void readout(const float* __restrict__ gsum,
                                               const float* __restrict__ gcnt,
                                               const float* __restrict__ Wr1,
                                               const float* __restrict__ br1,
                                               const float* __restrict__ Wr2,
                                               const float* __restrict__ br2,
                                               float* __restrict__ out) {
    const int g = blockIdx.x;
    const int j = threadIdx.x;   // 0..127
    const float inv = 1.0f / fmaxf(gcnt[g], 1.0f);
    const float* row = gsum + (size_t)g * 256;
    float acc = br1[j];
#pragma unroll 4
    for (int k = 0; k < 256; ++k) acc += (row[k] * inv) * Wr1[(size_t)k * 128 + j];
    // no ReLU on hidden in the reference
    __shared__ float red[128];
    red[j] = acc * Wr2[j];
    __syncthreads();
    for (int s = 64; s > 0; s >>= 1) {
        if (j < s) red[j] += red[j + s];
        __syncthreads();
    }
    if (j == 0) out[g] = red[0] + br2[0];
}

// ---------------------------------------------------------------------------
// Launcher
// ---------------------------------------------------------------------------
extern "C" void kernel_launch(void* const* d_in, const int* in_sizes, int n_in,
                              void* d_out, int out_size, void* d_ws, size_t ws_size,
                              hipStream_t stream) {
    const float* h   = (const float*)d_in[0];
    const int*   src = (const int*)d_in[1];
    const int*   dst = (const int*)d_in[2];
    const int*   gid = (const int*)d_in[3];
    const float* W1  = (const float*)d_in[4];
    const float* b1  = (const float*)d_in[5];
    const float* W2  = (const float*)d_in[6];
    const float* b2  = (const float*)d_in[7];
    const float* Wr1 = (const float*)d_in[8];
    const float* br1 = (const float*)d_in[9];
    const float* Wr2 = (const float*)d_in[10];
    const float* br2 = (const float*)d_in[11];
    float* out = (float*)d_out;

    const int N = in_sizes[3];   // 50000 (graph_ids length)
    const int E = in_sizes[1];   // 800000

    // Workspace layout (float elements, 64-elt aligned => 256B alignment)
    auto al64 = [](size_t x) { return (x + 63) & ~(size_t)63; };
    float* ws = (float*)d_ws;
    const size_t off_deg  = 0;
    const size_t off_agg1 = al64(off_deg + (size_t)N);
    const size_t off_h1   = al64(off_agg1 + (size_t)N * 128);   // also reused as h2
    const size_t off_agg2 = al64(off_h1 + (size_t)N * 256);
    const size_t off_gsum = al64(off_agg2 + (size_t)N * 256);
    const size_t off_gcnt = al64(off_gsum + (size_t)NUM_GRAPHS * 256);
    const size_t off_end  = off_gcnt + NUM_GRAPHS;
    (void)ws_size; (void)n_in; (void)off_end;

    // Zero all atomic accumulators (deg+agg1, agg2+gsum+gcnt); h1 region is
    // fully overwritten by the GEMM so it is skipped.
    hipMemsetAsync(ws + off_deg, 0, (off_h1 - off_deg) * sizeof(float), stream);
    hipMemsetAsync(ws + off_agg2, 0, (off_end - off_agg2) * sizeof(float), stream);

    float* deg  = ws + off_deg;
    float* agg1 = ws + off_agg1;
    float* h1   = ws + off_h1;     // layer-1 output, later reused for layer-2 output
    float* agg2 = ws + off_agg2;
    float* gsum = ws + off_gsum;
    float* gcnt = ws + off_gcnt;

    // 1) degree
    degree_count<<<(E + 255) / 256, 256, 0, stream>>>(dst, deg, E);

    // 2) aggregate layer 1 (128 feats)
    const long long tot1 = (long long)E * 32;
    edge_scatter_add<128><<<(int)((tot1 + 255) / 256), 256, 0, stream>>>(h, src, dst, agg1, tot1);

    // 3) GEMM1: relu((agg1/deg) @ W1 + b1) -> h1   [N,256]
    const int tiles = (N / 16) * 16;                 // N multiple of 16
    gemm_mean_bias_relu<128><<<(tiles + 7) / 8, 256, 0, stream>>>(agg1, deg, W1, b1, h1, N);

    // 4) aggregate layer 2 (256 feats)
    const long long tot2 = (long long)E * 64;
    edge_scatter_add<256><<<(int)((tot2 + 255) / 256), 256, 0, stream>>>(h1, src, dst, agg2, tot2);

    // 5) GEMM2: relu((agg2/deg) @ W2 + b2) -> h2 (reuses h1 buffer)
    gemm_mean_bias_relu<256><<<(tiles + 7) / 8, 256, 0, stream>>>(agg2, deg, W2, b2, h1, N);

    // 6) per-graph mean numerator/denominator
    const long long tot3 = (long long)N * 64;
    graph_accumulate<<<(int)((tot3 + 255) / 256), 256, 0, stream>>>(h1, gid, gsum, gcnt, tot3);

    // 7) readout head -> out[256]
    readout<<<NUM_GRAPHS, 128, 0, stream>>>(gsum, gcnt, Wr1, br1, Wr2, br2, out);
}